// RNN_51926154609260
// MI455X (gfx1250) — compile-verified
//
#include <hip/hip_runtime.h>
#include <hip/hip_bf16.h>
#include <math.h>

typedef float v2f __attribute__((ext_vector_type(2)));
typedef float v8f __attribute__((ext_vector_type(8)));

#define RNN_H   2048
#define SEQ     4096
#define OUTN    1000
#define SCAN_NB 64
#define SCAN_ROWS 32      // RNN_H / SCAN_NB
#define SCAN_THREADS 256

// ---------------------------------------------------------------------------
// C[M,N] = A[M,K] @ B[N,K]^T + bias[N]   (fp32, V_WMMA_F32_16X16X4_F32)
// One wave computes a 32x32 C block (2x2 grid of 16x16 WMMA tiles) so each
// k-step issues 4 WMMAs for 4 b64 fragment loads (1 load per WMMA).
// blockDim.x = 128 (4 waves, 32-col tiles along N => block covers 32x128).
// Fragment layouts per CDNA5 ISA 7.12.2:
//   A 16x4 : lanes 0-15 hold M=lane, {v0,v1}={K0,K1}; lanes 16-31 {K2,K3}
//   B 4x16 : lanes 0-15 hold N=lane, {v0,v1}={K0,K1}; lanes 16-31 {K2,K3}
//   C/D    : VGPR i -> M = i + 8*(lane>=16), N = lane&15
// Out-of-range N columns: B row index is clamped (safe load, garbage feeds
// only the unstored columns), stores are guarded.
// ---------------------------------------------------------------------------
__global__ void wmma_gemm_f32_nt(const float* __restrict__ A,
                                 const float* __restrict__ B,
                                 const float* __restrict__ bias,
                                 float* __restrict__ C,
                                 int M, int N, int K) {
  const int lane = threadIdx.x & 31;
  const int wave = threadIdx.x >> 5;
  const int tn   = blockIdx.x * 4 + wave;   // 32-column tile index
  const int tm   = blockIdx.y;              // 32-row tile index
  if (tn * 32 >= N) return;                 // wave-uniform: EXEC stays all-ones

  const int half = lane >> 4;               // 0: K0/K1, 1: K2/K3
  const int r    = lane & 15;
  const int m0   = tm * 32 + r;
  const int n0   = tn * 32 + r;
  const int n1   = n0 + 16;
  const int n0c  = (n0 < N) ? n0 : (N - 1); // clamp => branch-free loads
  const int n1c  = (n1 < N) ? n1 : (N - 1);

  const float* __restrict__ a0 = A + (size_t)m0 * K;
  const float* __restrict__ a1 = A + (size_t)(m0 + 16) * K;
  const float* __restrict__ b0 = B + (size_t)n0c * K;
  const float* __restrict__ b1 = B + (size_t)n1c * K;

  v8f c00 = {0.f,0.f,0.f,0.f,0.f,0.f,0.f,0.f};
  v8f c01 = c00, c10 = c00, c11 = c00;

  for (int k0 = 0; k0 < K; k0 += 4) {
    const int kb = k0 + half * 2;           // 8B-aligned => single b64 loads
    const v2f af0 = *(const v2f*)(a0 + kb);
    const v2f af1 = *(const v2f*)(a1 + kb);
    const v2f bf0 = *(const v2f*)(b0 + kb);
    const v2f bf1 = *(const v2f*)(b1 + kb);
    c00 = __builtin_amdgcn_wmma_f32_16x16x4_f32(false, af0, false, bf0, (short)0, c00, false, false);
    c01 = __builtin_amdgcn_wmma_f32_16x16x4_f32(false, af0, false, bf1, (short)0, c01, false, false);
    c10 = __builtin_amdgcn_wmma_f32_16x16x4_f32(false, af1, false, bf0, (short)0, c10, false, false);
    c11 = __builtin_amdgcn_wmma_f32_16x16x4_f32(false, af1, false, bf1, (short)0, c11, false, false);
  }

  const int rowbase0 = tm * 32 + half * 8;
  const int rowbase1 = rowbase0 + 16;
  if (n0 < N) {
    const float bv = bias[n0];
#pragma unroll
    for (int i = 0; i < 8; ++i) {
      C[(size_t)(rowbase0 + i) * N + n0] = c00[i] + bv;
      C[(size_t)(rowbase1 + i) * N + n0] = c10[i] + bv;
    }
  }
  if (n1 < N) {
    const float bv = bias[n1];
#pragma unroll
    for (int i = 0; i < 8; ++i) {
      C[(size_t)(rowbase0 + i) * N + n1] = c01[i] + bv;
      C[(size_t)(rowbase1 + i) * N + n1] = c11[i] + bv;
    }
  }
}

// ---------------------------------------------------------------------------
// Persistent RNN scan: 64 blocks, each owns 32 rows of W_hh staged into LDS
// (256 KB of the 320 KB WGP LDS) for the entire 4096-step scan. Per step:
// broadcast h_prev into LDS, 8 lanes per row dot-product from LDS, tanh,
// agent-scope store, device-wide arrive-counter barrier.
// ---------------------------------------------------------------------------
__global__ void rnn_scan(const float* __restrict__ W_hh,
                         const float* __restrict__ b_hh,
                         const float* __restrict__ xp,
                         float* __restrict__ hsbuf,   // (SEQ+1) x RNN_H, row0 = h0
                         unsigned* __restrict__ cnt) {
  extern __shared__ float smem[];
  float* sW = smem;                          // SCAN_ROWS * RNN_H floats (256 KB)
  float* sh = smem + SCAN_ROWS * RNN_H;      // RNN_H floats (8 KB)

  const int tid = threadIdx.x;
  const int rb  = blockIdx.x * SCAN_ROWS;

  // Stage this block's W_hh rows into LDS once (coalesced b128 loads).
  {
    const float4* src = (const float4*)(W_hh + (size_t)rb * RNN_H);
    float4* dst = (float4*)sW;
    const int n4 = SCAN_ROWS * RNN_H / 4;
    for (int i = tid; i < n4; i += SCAN_THREADS) dst[i] = src[i];
  }
  __syncthreads();

  const int row_local = tid >> 3;            // 0..31
  const int kpart     = tid & 7;             // 0..7, 256 k-values each
  const int row       = rb + row_local;
  const float bh      = b_hh[row];
  const float4* wv = (const float4*)(sW + (size_t)row_local * RNN_H + kpart * 256);

  for (int t = 0; t < SEQ; ++t) {
    // Broadcast h_prev (hsbuf row t) into LDS.
    {
      const float4* hsrc = (const float4*)(hsbuf + (size_t)t * RNN_H);
      float4* hdst = (float4*)sh;
      for (int i = tid; i < RNN_H / 4; i += SCAN_THREADS) hdst[i] = hsrc[i];
    }
    __syncthreads();

    const float4* hv = (const float4*)(sh + kpart * 256);
    float acc = 0.f;
#pragma unroll 8
    for (int i = 0; i < 64; ++i) {
      const float4 w = wv[i];
      const float4 h = hv[i];
      acc += w.x * h.x; acc += w.y * h.y;
      acc += w.z * h.z; acc += w.w * h.w;
    }
    // reduce the 8 k-partitions of this row (xor-reduce within groups of 8)
    acc += __shfl_xor(acc, 1, 8);
    acc += __shfl_xor(acc, 2, 8);
    acc += __shfl_xor(acc, 4, 8);

    if (kpart == 0) {
      const float hval = tanhf(acc + xp[(size_t)t * RNN_H + row] + bh);
      __hip_atomic_store(&hsbuf[(size_t)(t + 1) * RNN_H + row], hval,
                         __ATOMIC_RELAXED, __HIP_MEMORY_SCOPE_AGENT);
    }
    __threadfence();
    __syncthreads();

    // Device-wide barrier: monotonic arrive counter (reset by init kernel
    // each launch, so graph replays are deterministic).
    if (tid == 0) {
      __hip_atomic_fetch_add(cnt, 1u, __ATOMIC_RELEASE, __HIP_MEMORY_SCOPE_AGENT);
      const unsigned target = (unsigned)SCAN_NB * (unsigned)(t + 1);
      while (__hip_atomic_load(cnt, __ATOMIC_ACQUIRE, __HIP_MEMORY_SCOPE_AGENT) < target)
        __builtin_amdgcn_s_sleep(2);
    }
    __syncthreads();
  }
}

// Init: zero the barrier counter, seed hsbuf row 0 with h0, and emit the
// (unchanged) hidden state into the tail of d_out.
__global__ void rnn_init(const float* __restrict__ hidden,
                         float* __restrict__ hsbuf,
                         float* __restrict__ out_hidden,
                         unsigned* __restrict__ cnt) {
  const int i = blockIdx.x * blockDim.x + threadIdx.x;
  if (i < RNN_H) {
    const float v = hidden[i];
    hsbuf[i] = v;
    out_hidden[i] = v;
  }
  if (i == 0) *cnt = 0u;
}

extern "C" void kernel_launch(void* const* d_in, const int* in_sizes, int n_in,
                              void* d_out, int out_size, void* d_ws, size_t ws_size,
                              hipStream_t stream) {
  (void)in_sizes; (void)n_in; (void)out_size; (void)ws_size;

  const float* input  = (const float*)d_in[0];   // 4096 x 2048
  const float* hidden = (const float*)d_in[1];   // 1 x 2048
  const float* W_ih   = (const float*)d_in[2];   // 2048 x 2048
  const float* b_ih   = (const float*)d_in[3];   // 2048
  const float* W_hh   = (const float*)d_in[4];   // 2048 x 2048
  const float* b_hh   = (const float*)d_in[5];   // 2048
  const float* W_out  = (const float*)d_in[6];   // 1000 x 2048
  const float* b_out  = (const float*)d_in[7];   // 1000

  float* out = (float*)d_out;                    // [2048*1000 output | 2048 hidden]

  // Workspace layout
  float*    xp    = (float*)d_ws;                          // SEQ x H
  float*    hsbuf = xp + (size_t)SEQ * RNN_H;              // (SEQ+1) x H
  unsigned* cnt   = (unsigned*)(hsbuf + (size_t)(SEQ + 1) * RNN_H);

  // 0) init barrier counter, h0, and output hidden copy
  rnn_init<<<(RNN_H + 255) / 256, 256, 0, stream>>>(hidden, hsbuf,
                                                    out + (size_t)RNN_H * OUTN, cnt);

  // 1) xp = input @ W_ih^T + b_ih   (M=4096, N=2048, K=2048)
  {
    dim3 grid(RNN_H / 128, SEQ / 32);        // 16 x 128
    wmma_gemm_f32_nt<<<grid, 128, 0, stream>>>(input, W_ih, b_ih, xp,
                                               SEQ, RNN_H, RNN_H);
  }

  // 2) sequential scan (persistent kernel, W_hh resident in LDS)
  {
    const size_t lds = (size_t)(SCAN_ROWS * RNN_H + RNN_H) * sizeof(float); // 264 KB
    rnn_scan<<<SCAN_NB, SCAN_THREADS, lds, stream>>>(W_hh, b_hh, xp, hsbuf, cnt);
  }

  // 3) output = hs[-2048:] @ W_out^T + b_out   (M=2048, N=1000, K=2048)
  {
    const float* A2 = hsbuf + (size_t)(SEQ - RNN_H + 1) * RNN_H; // rows 2049..4096
    dim3 grid((OUTN + 127) / 128, RNN_H / 32);                   // 8 x 64
    wmma_gemm_f32_nt<<<grid, 128, 0, stream>>>(A2, W_out, b_out, out,
                                               RNN_H, OUTN, RNN_H);
  }
}